// SSIM_3350074491443
// MI455X (gfx1250) — compile-verified
//
#include <hip/hip_runtime.h>

// ---------------------------------------------------------------------------
// SSIM on MI455X (gfx1250, wave32).
// Separable 11-tap Gaussian conv as banded matmuls on the FP32 WMMA pipe:
//   Out(16x16) = Wv(16x28) * [ In(26x28) * Wh(28x16) ]
// per field (x, y, x^2, y^2, x*y), one wave per 16x16 output tile.
// Image tiles staged global->LDS with GLOBAL_LOAD_ASYNC_TO_LDS (ASYNCcnt).
// ---------------------------------------------------------------------------

typedef __attribute__((ext_vector_type(2))) float v2f;
typedef __attribute__((ext_vector_type(8))) float v8f;

#define IMG_H 512
#define IMG_W 512
#define NPLANE 48                       // B*C = 16*3
#define TILE 16
#define NTILES (NPLANE * 32 * 32)       // 49152
#define WAVES_PER_BLOCK 8
#define NBLOCKS (NTILES / WAVES_PER_BLOCK)  // 6144
#define INR 26                          // halo rows
#define INC 28                          // halo cols padded to K multiple of 4
#define TSZ (INR * INC)                 // 728 floats per staged tile
#define MIDSZ (TILE * INC)              // 448: midT[col][row], 16 x 28
#define LDS_PER_WAVE (2 * TSZ + MIDSZ)  // 1904 floats = 7616 B

// 11-tap Gaussian, sigma=1.5, centered at 5.5 (torch-faithful), normalized.
__constant__ float G11[11] = {
    3.2030e-04f, 2.9557e-03f, 1.7488e-02f, 6.6343e-02f, 1.6137e-01f,
    2.5169e-01f, 2.5169e-01f, 1.6137e-01f, 6.6343e-02f, 1.7488e-02f,
    2.9557e-03f};

__device__ __forceinline__ float gtap(int d) {
    return (d >= 0 && d < 11) ? G11[d] : 0.0f;
}

// ---- async global->LDS copy (CDNA5 path), 4 bytes per lane ----------------
__device__ __forceinline__ void async_g2l_b32(float* ldst, const float* gsrc) {
#if __has_builtin(__builtin_amdgcn_global_load_async_to_lds_b32)
    typedef __attribute__((address_space(1))) int gint_t;   // non-const per builtin proto
    typedef __attribute__((address_space(3))) int lint_t;
    __builtin_amdgcn_global_load_async_to_lds_b32(
        (gint_t*)(unsigned long long)gsrc,
        (lint_t*)(unsigned)(unsigned long long)ldst, 0, 0);
#else
    const unsigned l = (unsigned)(unsigned long long)ldst;   // LDS byte offset
    const unsigned long long g = (unsigned long long)gsrc;   // flat global addr
    asm volatile("global_load_async_to_lds_b32 %0, %1, off"
                 :: "v"(l), "v"(g)
                 : "memory");
#endif
}

__device__ __forceinline__ void wait_async0() {
#if __has_builtin(__builtin_amdgcn_s_wait_asynccnt)
    __builtin_amdgcn_s_wait_asynccnt(0);
    asm volatile("" ::: "memory");
#else
    asm volatile("s_wait_asynccnt 0x0" ::: "memory");
#endif
}

__device__ __forceinline__ v8f wmma4(v2f a, v2f b, v8f c) {
    return __builtin_amdgcn_wmma_f32_16x16x4_f32(false, a, false, b, (short)0,
                                                 c, false, false);
}

// contiguous K-pair of the field value; r in [0,26), k even in [0,28)
__device__ __forceinline__ v2f fpair(const float* xt, const float* yt,
                                     int r, int k, int f) {
    const int idx = r * INC + k;
    const v2f x = *(const v2f*)(xt + idx);
    const v2f y = *(const v2f*)(yt + idx);
    switch (f) {
        case 0: return x;
        case 1: return y;
        case 2: return x * x;
        case 3: return y * y;
        default: return x * y;
    }
}

__global__ __launch_bounds__(256) void ssim_tiles(const float* __restrict__ img1,
                                                  const float* __restrict__ img2,
                                                  float* __restrict__ partial) {
    __shared__ __align__(16) float lds[LDS_PER_WAVE * WAVES_PER_BLOCK +
                                       WAVES_PER_BLOCK];

    const int tid  = threadIdx.x;
    const int wave = tid >> 5;
    const int lane = tid & 31;
    const int m    = lane & 15;   // fragment row/col index
    const int half = lane >> 4;   // K sub-block (0 -> K+0,1 ; 1 -> K+2,3)

    float* xt    = &lds[wave * LDS_PER_WAVE];
    float* yt    = xt + TSZ;
    float* midT  = yt + TSZ;                              // [col][row] 16x28
    float* wpart = &lds[LDS_PER_WAVE * WAVES_PER_BLOCK];

    const int tileId = blockIdx.x * WAVES_PER_BLOCK + wave;
    const int plane  = tileId >> 10;
    const int trc    = tileId & 1023;
    const int r0     = (trc >> 5) * TILE;
    const int c0     = (trc & 31) * TILE;

    const float* p1 = img1 + (size_t)plane * (IMG_H * IMG_W);
    const float* p2 = img2 + (size_t)plane * (IMG_H * IMG_W);

    // ---- stage 26x28 halo tiles into LDS: async loads for in-image cells,
    //      DS zero-stores for pad cells (disjoint addresses -> no hazard)
    for (int idx = lane; idx < TSZ; idx += 32) {
        const int r  = idx / INC;
        const int c  = idx - r * INC;
        const int gr = r0 - 5 + r;
        const int gc = c0 - 5 + c;
        const bool inb = (c < INR) & (gr >= 0) & (gr < IMG_H) &
                         (gc >= 0) & (gc < IMG_W);
        if (inb) {
            const size_t o = (size_t)gr * IMG_W + gc;
            async_g2l_b32(&xt[idx], p1 + o);
            async_g2l_b32(&yt[idx], p2 + o);
        } else {
            xt[idx] = 0.0f;
            yt[idx] = 0.0f;
        }
    }
    // zero K-pad rows 26,27 of midT (column `lane`), never overwritten
    if (lane < TILE) {
        midT[lane * INC + 26] = 0.0f;
        midT[lane * INC + 27] = 0.0f;
    }

    // Banded-Gaussian fragments: same expression is the H-pass B-fragment
    // (Wh[k][n]=g[k-n]) and the V-pass A-fragment (Wv[m][k]=g[k-m]).
    v2f wfrag[7];
#pragma unroll
    for (int kk = 0; kk < 7; ++kk) {
        const int k = kk * 4 + 2 * half;
        wfrag[kk].x = gtap(k - m);
        wfrag[kk].y = gtap(k + 1 - m);
    }

    wait_async0();   // this wave's tile data resident in LDS

    v8f acc[5];
#pragma unroll
    for (int f = 0; f < 5; ++f) {
        // ---- H-pass: Mid(26x16) = Field(26x28) x Wh(28x16)
        //      top block rows 0..15, bottom block rows 10..25 (overlap rows
        //      10..15 recompute bit-identical values -> guard-free stores)
        v8f dt = {};
        v8f db = {};
#pragma unroll
        for (int kk = 0; kk < 7; ++kk) {
            const int k  = kk * 4 + 2 * half;
            const v2f at = fpair(xt, yt, m, k, f);
            const v2f ab = fpair(xt, yt, 10 + m, k, f);
            dt = wmma4(at, wfrag[kk], dt);
            db = wmma4(ab, wfrag[kk], db);
        }
        // D-fragment -> midT (transposed: 8 consecutive floats per lane)
#pragma unroll
        for (int v = 0; v < 8; ++v) {
            midT[m * INC + (v + 8 * half)]      = dt[v];
            midT[m * INC + (10 + v + 8 * half)] = db[v];
        }

        // ---- V-pass: Out(16x16) = Wv(16x28) x Mid(28x16)
        v8f dv = {};
#pragma unroll
        for (int kk = 0; kk < 7; ++kk) {
            const int k = kk * 4 + 2 * half;
            const v2f b = *(const v2f*)(midT + m * INC + k);
            dv = wmma4(wfrag[kk], b, dv);
        }
        acc[f] = dv;
        // per-wave LDS, DS ops in-order per wave: no barrier needed
    }

    // ---- pointwise SSIM map on the 8 output pixels held by this lane
    const float C1v = 0.01f * 0.01f;
    const float C2v = 0.03f * 0.03f;
    float s = 0.0f;
#pragma unroll
    for (int v = 0; v < 8; ++v) {
        const float mu1  = acc[0][v];
        const float mu2  = acc[1][v];
        const float mu1s = mu1 * mu1;
        const float mu2s = mu2 * mu2;
        const float m12  = mu1 * mu2;
        const float s1   = acc[2][v] - mu1s;
        const float s2   = acc[3][v] - mu2s;
        const float s12  = acc[4][v] - m12;
        const float num  = (2.0f * m12 + C1v) * (2.0f * s12 + C2v);
        const float den  = (mu1s + mu2s + C1v) * (s1 + s2 + C2v);
        s += num / den;
    }

    // ---- wave32 reduction, then per-block partial (deterministic)
#pragma unroll
    for (int off = 16; off > 0; off >>= 1) s += __shfl_xor(s, off, 32);
    if (lane == 0) wpart[wave] = s;
    __syncthreads();
    if (tid == 0) {
        float t = 0.0f;
#pragma unroll
        for (int wv = 0; wv < WAVES_PER_BLOCK; ++wv) t += wpart[wv];
        partial[blockIdx.x] = t;
    }
}

__global__ __launch_bounds__(256) void ssim_reduce(const float* __restrict__ partial,
                                                   float* __restrict__ out, int n) {
    __shared__ float sm[256];
    float s = 0.0f;
    for (int i = threadIdx.x; i < n; i += 256) s += partial[i];
    sm[threadIdx.x] = s;
    __syncthreads();
    for (int st = 128; st > 0; st >>= 1) {
        if (threadIdx.x < st) sm[threadIdx.x] += sm[threadIdx.x + st];
        __syncthreads();
    }
    if (threadIdx.x == 0)
        out[0] = sm[0] * (1.0f / (16.0f * 3.0f * 512.0f * 512.0f));
}

extern "C" void kernel_launch(void* const* d_in, const int* in_sizes, int n_in,
                              void* d_out, int out_size, void* d_ws, size_t ws_size,
                              hipStream_t stream) {
    const float* img1 = (const float*)d_in[0];
    const float* img2 = (const float*)d_in[1];
    float* partial = (float*)d_ws;   // NBLOCKS floats = 24 KB scratch

    ssim_tiles<<<NBLOCKS, 256, 0, stream>>>(img1, img2, partial);
    ssim_reduce<<<1, 256, 0, stream>>>(partial, (float*)d_out, NBLOCKS);
}